// aten_histc_out_36687610643116
// MI455X (gfx1250) — compile-verified
//
#include <hip/hip_runtime.h>
#include <stdint.h>

// ---------------------------------------------------------------------------
// histc(64M f32 -> 256 bins) for MI455X (gfx1250).
// Memory-bound: 256 MB read @ 23.3 TB/s => ~11 us floor. CDNA5 path: async
// global->LDS DMA (global_load_async_to_lds_b128 / s_wait_asynccnt), double
// buffered 32 B/thread/iter, + lane-sliced bank-conflict-free LDS histogram
// (wave32: bank of hist[bin*32+lane] is unique per lane).
// ---------------------------------------------------------------------------

#define BINS        256
#define THREADS     256          // 8 wave32s per workgroup
#define SLICES      32           // one histogram slice per lane of a wave32
#define HIST_BLOCKS 2048         // 128 elements/thread at n = 2^26

// LDS byte-offset of a __shared__ object (generic -> addrspace(3) -> int).
__device__ __forceinline__ unsigned lds_off_of(const void* p) {
    return (unsigned)(unsigned long long)(__attribute__((address_space(3))) const char*)p;
}

// CDNA5 async copy: 16 bytes global -> LDS, tracked by ASYNCcnt (no VGPR dest).
__device__ __forceinline__ void async_load_b128(unsigned lds_byte_off, const float* gaddr) {
    asm volatile("global_load_async_to_lds_b128 %0, %1, off"
                 :: "v"(lds_byte_off), "v"(gaddr)
                 : "memory");
}

__global__ void zero_ws_kernel(unsigned* __restrict__ ws) {
    ws[threadIdx.x] = 0u;
}

__global__ __launch_bounds__(THREADS)
void histc_kernel(const float* __restrict__ x, unsigned* __restrict__ ws, long long n) {
    __shared__ unsigned hist[BINS * SLICES];     // 32 KB, conflict-free slices
    __shared__ float4   stage[2][THREADS][2];    // 16 KB double buffer (32 B/thread)

    const int tid  = threadIdx.x;
    const int lane = tid & (SLICES - 1);

    // Vectorized zero-init: 8 x ds_store_b128 per thread.
    uint4* h4i = (uint4*)hist;
    #pragma unroll
    for (int k = 0; k < (BINS * SLICES) / (4 * THREADS); ++k)
        h4i[k * THREADS + tid] = make_uint4(0u, 0u, 0u, 0u);
    __syncthreads();

    const float lo    = -4.0f;
    const float hi    =  4.0f;
    const float scale = 32.0f;   // BINS / (HI - LO), exact

    // 32-bit chunk indexing: one chunk = 8 floats = 32 B.
    const unsigned nc    = (unsigned)(n >> 3);
    const unsigned total = gridDim.x * (unsigned)THREADS;
    const unsigned ncu   = (nc / total) * total;               // uniform region
    unsigned c = blockIdx.x * (unsigned)THREADS + (unsigned)tid;

    const unsigned sbase = lds_off_of(&stage[0][0][0]);
    const unsigned soff0 = sbase + (unsigned)tid * 32u;
    const unsigned soff1 = soff0 + THREADS * 32u;

    if (c < ncu) {
        async_load_b128(soff0,       x + (size_t)c * 8);       // prologue pair
        async_load_b128(soff0 + 16u, x + (size_t)c * 8 + 4);
        int p = 0;
        for (; c < ncu; c += total) {
            const unsigned cn = c + total;
            if (cn < ncu) {                                    // prefetch pair
                const unsigned so = p ? soff0 : soff1;
                async_load_b128(so,       x + (size_t)cn * 8);
                async_load_b128(so + 16u, x + (size_t)cn * 8 + 4);
                asm volatile("s_wait_asynccnt 0x2" ::: "memory");  // old pair done
            } else {
                asm volatile("s_wait_asynccnt 0x0" ::: "memory");  // drain
            }
            const float4 v0 = stage[p][tid][0];
            const float4 v1 = stage[p][tid][1];
            const float f[8] = {v0.x, v0.y, v0.z, v0.w, v1.x, v1.y, v1.z, v1.w};
            #pragma unroll
            for (int e = 0; e < 8; ++e) {
                const float fe = f[e];
                if (fe >= lo && fe <= hi) {                    // NaN rejected
                    int b = (int)floorf((fe - lo) * scale);    // sub+mul+floor (no fma)
                    b = (b > BINS - 1) ? (BINS - 1) : b;       // x == HI -> last bin
                    atomicAdd(&hist[b * SLICES + lane], 1u);   // ds_add_u32
                }
            }
            p ^= 1;
        }
    }

    // Tail (non-uniform remainder): plain global loads, same binning.
    for (long long j = (long long)ncu * 8 + (long long)blockIdx.x * THREADS + tid;
         j < n; j += (long long)total) {
        const float fe = x[j];
        if (fe >= lo && fe <= hi) {
            int b = (int)floorf((fe - lo) * scale);
            b = (b > BINS - 1) ? (BINS - 1) : b;
            atomicAdd(&hist[b * SLICES + lane], 1u);
        }
    }

    __syncthreads();

    // Per-block reduce: thread t owns bin t; 8 x ds_load_b128 + horizontal add.
    const uint4* h4 = (const uint4*)hist;
    unsigned sum = 0u;
    #pragma unroll
    for (int k = 0; k < SLICES / 4; ++k) {
        const uint4 u = h4[tid * (SLICES / 4) + k];
        sum += u.x + u.y + u.z + u.w;
    }
    atomicAdd(&ws[tid], sum);                                  // global_atomic_add_u32
}

__global__ void finalize_kernel(const unsigned* __restrict__ ws,
                                float* __restrict__ out, int out_size) {
    const int t = threadIdx.x;
    const float v = (float)ws[t];
    out[t] = v;                                  // first tuple element
    if (out_size >= 2 * BINS) out[BINS + t] = v; // second tuple element (h, h)
}

extern "C" void kernel_launch(void* const* d_in, const int* in_sizes, int n_in,
                              void* d_out, int out_size, void* d_ws, size_t ws_size,
                              hipStream_t stream) {
    const float* x = (const float*)d_in[0];
    unsigned* ws   = (unsigned*)d_ws;            // 256 u32 partial-count bins
    float* out     = (float*)d_out;
    const long long n = (long long)in_sizes[0];

    zero_ws_kernel<<<1, BINS, 0, stream>>>(ws);
    histc_kernel<<<HIST_BLOCKS, THREADS, 0, stream>>>(x, ws, n);
    finalize_kernel<<<1, BINS, 0, stream>>>(ws, out, out_size);
}